// TVA_block_att_22136261444464
// MI455X (gfx1250) — compile-verified
//
#include <hip/hip_runtime.h>
#include <hip/hip_bf16.h>

#define B_DIM 32
#define N_DIM 307
#define S_DIM 16
#define L_DIM 168
#define LN_EPS 1e-5f

typedef __attribute__((ext_vector_type(2))) float v2f;
typedef __attribute__((ext_vector_type(8))) float v8f;

static constexpr size_t E  = (size_t)B_DIM * N_DIM * S_DIM * L_DIM;  // 26,406,912
static constexpr size_t E2 = (size_t)L_DIM * L_DIM * S_DIM;          // 451,584

__device__ __forceinline__ v8f wmma4(v2f a, v2f b, v8f c) {
  // D = A(16x4 f32) * B(4x16 f32) + C(16x16 f32)
  return __builtin_amdgcn_wmma_f32_16x16x4_f32(false, a, false, b, (short)0, c,
                                               false, false);
}

__device__ __forceinline__ void lds_fence() {
  asm volatile("s_wait_dscnt 0x0" ::: "memory");
}

// ---------------------------------------------------------------------------
// WMMA GEMM, 16(M) x 64(N) per wave:  out[r,n] = sum_k A[r,k]*W[n*K+k] + bias[n]
// MODE 0: A[r,k] = Abuf[r*K + k]                      (contiguous rows)
// MODE 1: rows r=(b,l,s); A[r,k] = H[b, n=k, s, l]    (stride S*L gather)
// Branchless: partial columns load from clamped addresses (garbage only feeds
// masked-out D columns); K-tail zeroes the A fragment via cndmask.
// M must be a multiple of 16.
// ---------------------------------------------------------------------------
template <int MODE>
__global__ __launch_bounds__(256)
void wmma_gemm_kernel(const float* __restrict__ Abuf,
                      const float* __restrict__ W,
                      const float* __restrict__ bias,
                      float* __restrict__ out,
                      int K, int Nout, int totalWaves, int nGroups) {
  int wave = blockIdx.x * (blockDim.x >> 5) + (threadIdx.x >> 5);
  if (wave >= totalWaves) return;        // wave-uniform: EXEC stays all-ones
  int lane = threadIdx.x & 31;
  int tm = wave / nGroups, tg = wave % nGroups;
  int half = lane >> 4;                  // 0: K pair {0,1}, 1: K pair {2,3}
  int kb = half << 1;
  int mrow = (tm << 4) + (lane & 15);    // A row this lane supplies

  size_t abase, akstr;
  if (MODE == 0) {
    abase = (size_t)mrow * K;
    akstr = 1;
  } else {
    int bb = mrow / (L_DIM * S_DIM);
    int t  = mrow % (L_DIM * S_DIM);
    int l = t / S_DIM, s = t % S_DIM;
    abase = (size_t)bb * N_DIM * S_DIM * L_DIM + (size_t)s * L_DIM + l;
    akstr = (size_t)S_DIM * L_DIM;
  }

  int col0 = ((tg << 2) + 0) * 16 + (lane & 15);
  int col1 = ((tg << 2) + 1) * 16 + (lane & 15);
  int col2 = ((tg << 2) + 2) * 16 + (lane & 15);
  int col3 = ((tg << 2) + 3) * 16 + (lane & 15);
  size_t wr0 = (size_t)(col0 < Nout ? col0 : Nout - 1) * K;
  size_t wr1 = (size_t)(col1 < Nout ? col1 : Nout - 1) * K;
  size_t wr2 = (size_t)(col2 < Nout ? col2 : Nout - 1) * K;
  size_t wr3 = (size_t)(col3 < Nout ? col3 : Nout - 1) * K;

  v8f c0 = {}, c1 = {}, c2 = {}, c3 = {};
  int K4 = K & ~3;
  for (int kk = 0; kk < K4; kk += 4) {
    int k0 = kk + kb;
    v2f a;
    if (MODE == 0) {
      a = *(const v2f*)(Abuf + abase + k0);          // even offset: b64 load
    } else {
      a.x = Abuf[abase + (size_t)k0 * akstr];
      a.y = Abuf[abase + (size_t)(k0 + 1) * akstr];
    }
    if (MODE == 0) {
      v2f b0 = *(const v2f*)(W + wr0 + k0);
      v2f b1 = *(const v2f*)(W + wr1 + k0);
      v2f b2 = *(const v2f*)(W + wr2 + k0);
      v2f b3 = *(const v2f*)(W + wr3 + k0);
      c0 = wmma4(a, b0, c0);
      c1 = wmma4(a, b1, c1);
      c2 = wmma4(a, b2, c2);
      c3 = wmma4(a, b3, c3);
    } else {                                          // K=307: odd strides
      v2f b0, b1, b2, b3;
      b0.x = W[wr0 + k0]; b0.y = W[wr0 + k0 + 1];
      b1.x = W[wr1 + k0]; b1.y = W[wr1 + k0 + 1];
      b2.x = W[wr2 + k0]; b2.y = W[wr2 + k0 + 1];
      b3.x = W[wr3 + k0]; b3.y = W[wr3 + k0 + 1];
      c0 = wmma4(a, b0, c0);
      c1 = wmma4(a, b1, c1);
      c2 = wmma4(a, b2, c2);
      c3 = wmma4(a, b3, c3);
    }
  }
  if (K4 < K) {  // tail (K=307): zero A beyond K; B reads clamped addresses
    int k0 = K4 + kb;
    int kx = k0 < K ? k0 : K - 1;
    int ky = (k0 + 1) < K ? (k0 + 1) : K - 1;
    float ax = Abuf[abase + (size_t)kx * akstr];
    float ay = Abuf[abase + (size_t)ky * akstr];
    v2f a;
    a.x = (k0 < K) ? ax : 0.f;
    a.y = (k0 + 1 < K) ? ay : 0.f;
    v2f b0, b1, b2, b3;
    b0.x = W[wr0 + kx]; b0.y = W[wr0 + ky];
    b1.x = W[wr1 + kx]; b1.y = W[wr1 + ky];
    b2.x = W[wr2 + kx]; b2.y = W[wr2 + ky];
    b3.x = W[wr3 + kx]; b3.y = W[wr3 + ky];
    c0 = wmma4(a, b0, c0);
    c1 = wmma4(a, b1, c1);
    c2 = wmma4(a, b2, c2);
    c3 = wmma4(a, b3, c3);
  }

  int rbase = (tm << 4) + (half << 3);   // C/D layout: M = r + 8*half
#define STORE_TILE(ACC, COL)                                          \
  if ((COL) < Nout) {                                                 \
    float bv = bias[(COL)];                                           \
    for (int r = 0; r < 8; ++r)                                       \
      out[(size_t)(rbase + r) * Nout + (COL)] = (ACC)[r] + bv;        \
  }
  STORE_TILE(c0, col0)
  STORE_TILE(c1, col1)
  STORE_TILE(c2, col2)
  STORE_TILE(c3, col3)
#undef STORE_TILE
}

// ---------------------------------------------------------------------------
// Time/cross attention: per (b,n) slice of [S=16, L=168]:
//   P = softmax(Q Kt / sqrt(2));  res = (P V)*wo_sum + bo;  out = LN(H + res)
// One wave per slice, 4 waves per block.
// ---------------------------------------------------------------------------
__global__ __launch_bounds__(128)
void attn_ln_kernel(const float* __restrict__ Q,
                    const float* __restrict__ Kb,
                    const float* __restrict__ V,
                    const float* __restrict__ H,
                    const float* __restrict__ wo,
                    const float* __restrict__ bo,
                    const float* __restrict__ g,
                    const float* __restrict__ beta,
                    float* __restrict__ out, int numSlices) {
  __shared__ float Pst[4][16 * 16];
  __shared__ float Yst[4][16 * L_DIM];
  __shared__ float Stat[4][32];
  int wv = threadIdx.x >> 5;
  int lane = threadIdx.x & 31;
  int slice = blockIdx.x * 4 + wv;
  if (slice >= numSlices) return;
  size_t base = (size_t)slice * S_DIM * L_DIM;
  const float* q = Q + base;
  const float* k = Kb + base;
  const float* v = V + base;
  const float* h = H + base;
  float wos = 0.f;
  for (int i = 0; i < 8; ++i) wos += wo[i];
  float bo0 = bo[0];
  const float inv_scale = 0.70710678118654752f;  // 1/sqrt(S/NUM_HEAD)

  int half = lane >> 4;
  int kb = half << 1;
  int m = lane & 15;

  // scores = Q @ K^T  (16x16, contract L); both fragment pairs contiguous
  v8f c = {};
  for (int kk = 0; kk < L_DIM; kk += 4) {
    v2f a = *(const v2f*)(q + m * L_DIM + kk + kb);
    v2f b = *(const v2f*)(k + m * L_DIM + kk + kb);  // B[k',n]=K[n,kk+k']
    c = wmma4(a, b, c);
  }
  // softmax rows (each row lives across the 16 lanes of a half-wave)
  for (int r = 0; r < 8; ++r) {
    float s = c[r] * inv_scale;
    float mx = s;
    for (int off = 8; off >= 1; off >>= 1) mx = fmaxf(mx, __shfl_xor(mx, off, 32));
    float e = __expf(s - mx);
    float sm = e;
    for (int off = 8; off >= 1; off >>= 1) sm += __shfl_xor(sm, off, 32);
    Pst[wv][(r + (half << 3)) * 16 + (lane & 15)] = e / sm;
  }
  lds_fence();

  // out = P @ V, fused epilogue into LDS for LayerNorm
  for (int tn = 0; tn < (L_DIM + 15) / 16; ++tn) {
    int col = (tn << 4) + (lane & 15);
    int colc = col < L_DIM ? col : L_DIM - 1;   // clamped; masked at store
    v8f d = {};
    for (int kk = 0; kk < 16; kk += 4) {
      v2f a = *(const v2f*)(&Pst[wv][m * 16 + kk + kb]);
      v2f b;
      b.x = v[(kk + kb) * L_DIM + colc];
      b.y = v[(kk + kb + 1) * L_DIM + colc];
      d = wmma4(a, b, d);
    }
    if (col < L_DIM) {
      for (int r = 0; r < 8; ++r) {
        int row = r + (half << 3);
        Yst[wv][row * L_DIM + col] = h[row * L_DIM + col] + d[r] * wos + bo0;
      }
    }
  }
  lds_fence();

  // LayerNorm over L per row
  if (lane < 16) {
    float mean = 0.f;
    for (int j = 0; j < L_DIM; ++j) mean += Yst[wv][lane * L_DIM + j];
    mean /= (float)L_DIM;
    float var = 0.f;
    for (int j = 0; j < L_DIM; ++j) {
      float dd = Yst[wv][lane * L_DIM + j] - mean;
      var += dd * dd;
    }
    var /= (float)L_DIM;
    Stat[wv][lane] = mean;
    Stat[wv][16 + lane] = rsqrtf(var + LN_EPS);
  }
  lds_fence();
  for (int r = 0; r < 16; ++r) {
    float mean = Stat[wv][r], rs = Stat[wv][16 + r];
    for (int j = lane; j < L_DIM; j += 32) {
      out[base + r * L_DIM + j] =
          (Yst[wv][r * L_DIM + j] - mean) * rs * g[j] + beta[j];
    }
  }
}

// ---------------------------------------------------------------------------
// Space attention: per (b,l) slice of [S=16, N=307] (buffers in [B,L,S,N]):
//   res = softmax(Q Kt / sqrt(2)) V * wo_sum + bo; scatter back to [B,N,S,L].
// ---------------------------------------------------------------------------
__global__ __launch_bounds__(128)
void space_attn_kernel(const float* __restrict__ Q,
                       const float* __restrict__ Kb,
                       const float* __restrict__ V,
                       const float* __restrict__ wo,
                       const float* __restrict__ bo,
                       float* __restrict__ out, int numSlices) {
  __shared__ float Pst[4][16 * 16];
  int wv = threadIdx.x >> 5;
  int lane = threadIdx.x & 31;
  int slice = blockIdx.x * 4 + wv;
  if (slice >= numSlices) return;
  int bb = slice / L_DIM, l = slice % L_DIM;
  size_t base = (size_t)slice * S_DIM * N_DIM;
  const float* q = Q + base;
  const float* k = Kb + base;
  const float* v = V + base;
  float wos = 0.f;
  for (int i = 0; i < 8; ++i) wos += wo[i];
  float bo0 = bo[0];
  const float inv_scale = 0.70710678118654752f;

  int half = lane >> 4;
  int kb = half << 1;
  int m = lane & 15;

  // scores: K = 307; main loop to 304, branchless tail zeroes A fragment
  v8f c = {};
  for (int kk = 0; kk < (N_DIM & ~3); kk += 4) {
    int k0 = kk + kb;
    v2f a, b;
    a.x = q[m * N_DIM + k0]; a.y = q[m * N_DIM + k0 + 1];
    b.x = k[m * N_DIM + k0]; b.y = k[m * N_DIM + k0 + 1];
    c = wmma4(a, b, c);
  }
  {
    int k0 = (N_DIM & ~3) + kb;
    int kx = k0 < N_DIM ? k0 : N_DIM - 1;
    int ky = (k0 + 1) < N_DIM ? (k0 + 1) : N_DIM - 1;
    float ax = q[m * N_DIM + kx], ay = q[m * N_DIM + ky];
    v2f a, b;
    a.x = (k0 < N_DIM) ? ax : 0.f;
    a.y = (k0 + 1 < N_DIM) ? ay : 0.f;
    b.x = k[m * N_DIM + kx];
    b.y = k[m * N_DIM + ky];
    c = wmma4(a, b, c);
  }
  for (int r = 0; r < 8; ++r) {
    float s = c[r] * inv_scale;
    float mx = s;
    for (int off = 8; off >= 1; off >>= 1) mx = fmaxf(mx, __shfl_xor(mx, off, 32));
    float e = __expf(s - mx);
    float sm = e;
    for (int off = 8; off >= 1; off >>= 1) sm += __shfl_xor(sm, off, 32);
    Pst[wv][(r + (half << 3)) * 16 + (lane & 15)] = e / sm;
  }
  lds_fence();

  for (int tn = 0; tn < (N_DIM + 15) / 16; ++tn) {
    int col = (tn << 4) + (lane & 15);
    int colc = col < N_DIM ? col : N_DIM - 1;   // clamped; masked at store
    v8f d = {};
    for (int kk = 0; kk < 16; kk += 4) {
      v2f a = *(const v2f*)(&Pst[wv][m * 16 + kk + kb]);
      v2f b;
      b.x = v[(kk + kb) * N_DIM + colc];
      b.y = v[(kk + kb + 1) * N_DIM + colc];
      d = wmma4(a, b, d);
    }
    if (col < N_DIM) {
      for (int r = 0; r < 8; ++r) {
        int s = r + (half << 3);
        // transpose back: out[b, n=col, s, l]
        out[(((size_t)bb * N_DIM + col) * S_DIM + s) * L_DIM + l] =
            d[r] * wos + bo0;
      }
    }
  }
}

// FW[m, s*L + l] = f_w[m, l, s]  so the final einsum is a plain GEMM
__global__ void transpose_fw_kernel(const float* __restrict__ fw,
                                    float* __restrict__ FW) {
  size_t idx = (size_t)blockIdx.x * blockDim.x + threadIdx.x;
  if (idx >= E2) return;
  int m = (int)(idx / (S_DIM * L_DIM));
  int rem = (int)(idx % (S_DIM * L_DIM));
  int s = rem / L_DIM, l = rem % L_DIM;
  FW[idx] = fw[((size_t)m * L_DIM + l) * S_DIM + s];
}

extern "C" void kernel_launch(void* const* d_in, const int* in_sizes, int n_in,
                              void* d_out, int out_size, void* d_ws,
                              size_t ws_size, hipStream_t stream) {
  (void)in_sizes; (void)n_in; (void)out_size; (void)ws_size;
  const float* x      = (const float*)d_in[0];
  const float* t_wq   = (const float*)d_in[1];
  const float* t_bq   = (const float*)d_in[2];
  const float* t_wk   = (const float*)d_in[3];
  const float* t_bk   = (const float*)d_in[4];
  const float* t_wv   = (const float*)d_in[5];
  const float* t_bv   = (const float*)d_in[6];
  const float* t_wo   = (const float*)d_in[7];
  const float* t_bo   = (const float*)d_in[8];
  const float* t_g    = (const float*)d_in[9];
  const float* t_beta = (const float*)d_in[10];
  const float* c_wq   = (const float*)d_in[11];
  const float* c_bq   = (const float*)d_in[12];
  const float* c_wk   = (const float*)d_in[13];
  const float* c_bk   = (const float*)d_in[14];
  const float* c_wv   = (const float*)d_in[15];
  const float* c_bv   = (const float*)d_in[16];
  const float* c_wo   = (const float*)d_in[17];
  const float* c_bo   = (const float*)d_in[18];
  const float* c_g    = (const float*)d_in[19];
  const float* c_beta = (const float*)d_in[20];
  const float* s_wq   = (const float*)d_in[21];
  const float* s_bq   = (const float*)d_in[22];
  const float* s_wk   = (const float*)d_in[23];
  const float* s_bk   = (const float*)d_in[24];
  const float* s_wv   = (const float*)d_in[25];
  const float* s_bv   = (const float*)d_in[26];
  const float* s_wo   = (const float*)d_in[27];
  const float* s_bo   = (const float*)d_in[28];
  const float* f_w    = (const float*)d_in[29];
  const float* f_b    = (const float*)d_in[30];

  float* H  = (float*)d_ws;
  float* Qb = H + E;
  float* Kb = Qb + E;
  float* Vb = Kb + E;
  float* TA = Vb + E;
  float* SA = TA + E;
  float* FW = SA + E;

  hipMemcpyAsync(H, x, E * sizeof(float), hipMemcpyDeviceToDevice, stream);
  transpose_fw_kernel<<<(int)((E2 + 255) / 256), 256, 0, stream>>>(f_w, FW);

  auto gemm = [&](const float* A, const float* W, const float* bias, float* out,
                  int M, int K, int Nout, int mode) {
    int mT = M >> 4;
    int nGroups = ((Nout + 15) >> 4 + 0);
    nGroups = (((Nout + 15) >> 4) + 3) >> 2;
    int total = mT * nGroups;
    int blocks = (total + 7) / 8;  // 8 waves (256 threads) per block
    if (mode == 0)
      wmma_gemm_kernel<0><<<blocks, 256, 0, stream>>>(A, W, bias, out, K, Nout,
                                                      total, nGroups);
    else
      wmma_gemm_kernel<1><<<blocks, 256, 0, stream>>>(A, W, bias, out, K, Nout,
                                                      total, nGroups);
  };

  const int BNS = B_DIM * N_DIM * S_DIM;  // 157,184 rows
  const int BLS = B_DIM * L_DIM * S_DIM;  //  86,016 rows
  const int sliceT = B_DIM * N_DIM;       // 9,824 (b,n) slices
  const int sliceS = B_DIM * L_DIM;       // 5,376 (b,l) slices

  for (int layer = 0; layer < 2; ++layer) {
    // time attention
    gemm(H, t_wq, t_bq, Qb, BNS, L_DIM, L_DIM, 0);
    gemm(H, t_wk, t_bk, Kb, BNS, L_DIM, L_DIM, 0);
    gemm(H, t_wv, t_bv, Vb, BNS, L_DIM, L_DIM, 0);
    attn_ln_kernel<<<(sliceT + 3) / 4, 128, 0, stream>>>(
        Qb, Kb, Vb, H, t_wo, t_bo, t_g, t_beta, TA, sliceT);
    // space attention (projections gather the transpose directly)
    gemm(H, s_wq, s_bq, Qb, BLS, N_DIM, N_DIM, 1);
    gemm(H, s_wk, s_bk, Kb, BLS, N_DIM, N_DIM, 1);
    gemm(H, s_wv, s_bv, Vb, BLS, N_DIM, N_DIM, 1);
    space_attn_kernel<<<(sliceS + 3) / 4, 128, 0, stream>>>(
        Qb, Kb, Vb, s_wo, s_bo, SA, sliceS);
    // cross attention: q from space out, k/v from time out
    gemm(SA, c_wq, c_bq, Qb, BNS, L_DIM, L_DIM, 0);
    gemm(TA, c_wk, c_bk, Kb, BNS, L_DIM, L_DIM, 0);
    gemm(TA, c_wv, c_bv, Vb, BNS, L_DIM, L_DIM, 0);
    attn_ln_kernel<<<(sliceT + 3) / 4, 128, 0, stream>>>(
        Qb, Kb, Vb, TA, c_wo, c_bo, c_g, c_beta, H, sliceT);
  }
  // final Conv2d == GEMM over K = S*L with pre-transposed weights
  gemm(H, FW, f_b, (float*)d_out, B_DIM * N_DIM, S_DIM * L_DIM, L_DIM, 0);
}